// CrossAttention_15006615733765
// MI455X (gfx1250) — compile-verified
//
#include <hip/hip_runtime.h>
#include <hip/hip_bf16.h>

// Cross-attention with a single broadcast context vector per batch collapses
// exactly: softmax over identical keys is uniform, so attention output == V.
//   y[b,:] = (context[b] @ Wv) @ Wo + bo ;  out[b,c,h,w] = y[b,c]
// Store-bound: ~19 MB broadcast write at 23.3 TB/s  =>  ~0.9 us floor.

#define BATCH 4
#define DIM   512
#define CTX   768
#define NSPAT (48 * 48)   // 2304 spatial positions

typedef __attribute__((ext_vector_type(2))) float v2f;
typedef __attribute__((ext_vector_type(8))) float v8f;

// D(16 x 512) = A(16 x K) * W(K x 512) [+ bias]; A rows >= BATCH are zero-pad.
// One wave (32 lanes) per 16-column tile; V_WMMA_F32_16X16X4_F32 per K-step.
// Uniform control flow: padding rows are handled with a per-lane multiplicative
// mask on an in-bounds (clamped) row load — no EXEC-divergent branches in the
// hot loop, so EXEC is trivially all-ones at every WMMA.
__global__ void __launch_bounds__(32)
gemm16_wmma_f32(const float* __restrict__ A, int lda,
                const float* __restrict__ W,       // [K x 512] row-major
                const float* __restrict__ bias,    // nullptr or [512]
                float* __restrict__ Out,           // [BATCH x 512]
                int K)
{
    const int lane = threadIdx.x;     // 0..31
    const int half = lane >> 4;       // 0 or 1 (selects K pair within K4 step)
    const int nl   = lane & 15;
    const int col  = blockIdx.x * 16 + nl;   // N index of this lane
    const int m    = nl;                     // A row of this lane

    // Zero-pad rows >= BATCH without divergence: clamp row, mask value.
    const int   mrow = (m < BATCH) ? m : 0;
    const float msk  = (m < BATCH) ? 1.0f : 0.0f;
    const float* __restrict__ Arow = A + (size_t)mrow * lda;
    const float* __restrict__ Wcol = W + col;

    v8f acc = {};
    #pragma unroll 4
    for (int k0 = 0; k0 < K; k0 += 4) {
        const int ka = k0 + 2 * half;            // even => 8B-aligned pair
        v2f a = *(const v2f*)(Arow + ka);        // one global_load_b64
        a.x *= msk;
        a.y *= msk;
        v2f b;
        b.x = Wcol[(size_t)ka * DIM];            // row ka
        b.y = Wcol[(size_t)(ka + 1) * DIM];      // row ka+1
        // v_wmma_f32_16x16x4_f32: (neg_a, A, neg_b, B, c_mod, C, reuse_a, reuse_b)
        acc = __builtin_amdgcn_wmma_f32_16x16x4_f32(
                  false, a, false, b, (short)0, acc, false, false);
    }

    // C/D layout: lanes 0-15 hold rows M=0..7 in acc[0..7]; we need rows 0..3.
    if (lane < 16) {
        const float bv = bias ? bias[col] : 0.0f;
        Out[0 * DIM + col] = acc[0] + bv;
        Out[1 * DIM + col] = acc[1] + bv;
        Out[2 * DIM + col] = acc[2] + bv;
        Out[3 * DIM + col] = acc[3] + bv;
    }
}

// out[b, c, n] = y[b*DIM + c] for all n; saturating b128 stores.
// NSPAT % 4 == 0, so each float4 lies in a single (b,c) row: row = i / 576.
__global__ void __launch_bounds__(256)
broadcast_rows(const float* __restrict__ y, float* __restrict__ out, int total4)
{
    const int i = blockIdx.x * blockDim.x + threadIdx.x;
    if (i >= total4) return;
    const int row = i / (NSPAT / 4);            // 576 float4 per (b,c) row
    const float v = y[row];
    const float4 val = make_float4(v, v, v, v);
    reinterpret_cast<float4*>(out)[i] = val;
}

extern "C" void kernel_launch(void* const* d_in, const int* in_sizes, int n_in,
                              void* d_out, int out_size, void* d_ws, size_t ws_size,
                              hipStream_t stream)
{
    // setup_inputs order: x(0), context(1), Wq(2), Wk(3), Wv(4), Wo(5), bo(6)
    const float* ctx = (const float*)d_in[1];
    const float* Wv  = (const float*)d_in[4];
    const float* Wo  = (const float*)d_in[5];
    const float* bo  = (const float*)d_in[6];
    float* out = (float*)d_out;

    float* wsT = (float*)d_ws;            // T = context @ Wv   [4 x 512]
    float* wsY = wsT + BATCH * DIM;       // y = T @ Wo + bo    [4 x 512]

    // Stage 1: T = context(4x768) @ Wv(768x512)
    gemm16_wmma_f32<<<DIM / 16, 32, 0, stream>>>(ctx, CTX, Wv, nullptr, wsT, CTX);
    // Stage 2: y = T(4x512) @ Wo(512x512) + bo
    gemm16_wmma_f32<<<DIM / 16, 32, 0, stream>>>(wsT, DIM, Wo, bo, wsY, DIM);
    // Stage 3: broadcast y over all 48*48 positions (store-bound, ~19 MB)
    const int total4 = BATCH * DIM * NSPAT / 4;   // 1,179,648 float4 stores
    broadcast_rows<<<(total4 + 255) / 256, 256, 0, stream>>>(wsY, out, total4);
}